// GCE_25074019074051
// MI455X (gfx1250) — compile-verified
//
#include <hip/hip_runtime.h>

typedef __attribute__((ext_vector_type(2))) float v2f;
typedef __attribute__((ext_vector_type(8))) float v8f;

#define N_NODES    50000
#define IN_DIM     128
#define OUT_DIM    64
#define N_EDGES    1600000
#define N_GATHER   8192         // 4096 * 2
#define N_MTILES   3125         // 50000 / 16
#define PAIR_PITCH 160          // 128 floats/pair-row + 32-float skew (bank-conflict free)
#define N_PAIRS    64           // 128 K values -> 64 (k, k+1) pairs

// ---------------------------------------------------------------- init
__global__ void gce_init(float* __restrict__ acc, float* __restrict__ deg, int nAcc) {
    int i = blockIdx.x * blockDim.x + threadIdx.x;
    if (i < nAcc) acc[i] = 0.0f;
    if (i < N_NODES) deg[i] = 1.0f;   // self-loop pre-counted
}

// ---------------------------------------------------------------- degree scatter
__global__ void gce_degree(const long long* __restrict__ dst, float* __restrict__ deg) {
    int e = blockIdx.x * blockDim.x + threadIdx.x;
    if (e < N_EDGES) {
        int d = (int)dst[e];
        atomicAdd(&deg[d], 1.0f);
    }
}

// ---------------------------------------------------------------- deg -> rsqrt(deg)
__global__ void gce_dinv(float* __restrict__ deg) {
    int i = blockIdx.x * blockDim.x + threadIdx.x;
    if (i < N_NODES) {
        float d = deg[i];
        deg[i] = (d > 0.0f) ? rsqrtf(d) : 0.0f;
    }
}

// ---------------------------------------------------------------- xw = features @ W  (WMMA fp32 16x16x4)
__global__ __launch_bounds__(256)
void gce_gemm(const float* __restrict__ features, const float* __restrict__ W,
              float* __restrict__ xw) {
    // Pair-interleaved W in LDS: sW[p*PAIR_PITCH + n*2 + {0,1}] = W[2p][n], W[2p+1][n]
    // -> each B fragment is one ds_load_b64; 32-bank skew between consecutive pair-rows.
    __shared__ float sW[N_PAIRS * PAIR_PITCH];      // 40 KB of the 320 KB LDS
    for (int i = threadIdx.x; i < N_PAIRS * OUT_DIM; i += 256) {
        int p = i >> 6;                             // k-pair index 0..63
        int n = i & 63;
        v2f w;
        w.x = W[(p * 2    ) * OUT_DIM + n];
        w.y = W[(p * 2 + 1) * OUT_DIM + n];
        *(v2f*)(sW + p * PAIR_PITCH + n * 2) = w;
    }
    __syncthreads();

    const int lane = threadIdx.x & 31;
    const int wave = threadIdx.x >> 5;
    const int tile = blockIdx.x * 8 + wave;         // 8 waves per block
    if (tile >= N_MTILES) return;

    const int m0   = tile * 16;
    const int nl   = lane & 15;
    const int kOff = (lane >> 4) * 2;               // lanes 0-15: K+0/K+1, lanes 16-31: K+2/K+3
    const int rowA = m0 + nl;

    v8f c0 = {}, c1 = {}, c2 = {}, c3 = {};

    for (int k0 = 0; k0 < IN_DIM; k0 += 4) {
        // A fragment: 16x4 tile, row = rowA, K = k0+kOff .. k0+kOff+1 (8-byte aligned)
        v2f a = *(const v2f*)(features + (size_t)rowA * IN_DIM + k0 + kOff);

        const float* base = sW + ((k0 + kOff) >> 1) * PAIR_PITCH;   // kb is always even
        v2f b0 = *(const v2f*)(base + ( 0 + nl) * 2);
        v2f b1 = *(const v2f*)(base + (16 + nl) * 2);
        v2f b2 = *(const v2f*)(base + (32 + nl) * 2);
        v2f b3 = *(const v2f*)(base + (48 + nl) * 2);

        c0 = __builtin_amdgcn_wmma_f32_16x16x4_f32(false, a, false, b0, (short)0, c0, false, false);
        c1 = __builtin_amdgcn_wmma_f32_16x16x4_f32(false, a, false, b1, (short)0, c1, false, false);
        c2 = __builtin_amdgcn_wmma_f32_16x16x4_f32(false, a, false, b2, (short)0, c2, false, false);
        c3 = __builtin_amdgcn_wmma_f32_16x16x4_f32(false, a, false, b3, (short)0, c3, false, false);
    }

    // C/D layout: VGPR r -> M = r (lanes 0-15) / M = 8+r (lanes 16-31), N = lane&15
    const int rbase = m0 + ((lane >> 4) ? 8 : 0);
#pragma unroll
    for (int r = 0; r < 8; ++r) {
        float* row = xw + (size_t)(rbase + r) * OUT_DIM;
        row[ 0 + nl] = c0[r];
        row[16 + nl] = c1[r];
        row[32 + nl] = c2[r];
        row[48 + nl] = c3[r];
    }
}

// ---------------------------------------------------------------- edge aggregation (scatter-add)
__global__ void gce_edges(const long long* __restrict__ src, const long long* __restrict__ dst,
                          const float* __restrict__ xw, const float* __restrict__ dinv,
                          float* __restrict__ acc) {
    long long tid = (long long)blockIdx.x * blockDim.x + threadIdx.x;
    long long e   = tid >> 5;                       // one wave32 per edge
    if (e >= N_EDGES) return;
    const int lane = (int)(tid & 31);

    // e is wave-uniform: keep indices/norm in SGPRs
    const int s = __builtin_amdgcn_readfirstlane((int)src[e]);
    const int d = __builtin_amdgcn_readfirstlane((int)dst[e]);
    const float nrm = dinv[s] * dinv[d];

    v2f m = *(const v2f*)(xw + (size_t)s * OUT_DIM + lane * 2);
    float* dp = acc + (size_t)d * OUT_DIM + lane * 2;
    atomicAdd(dp    , m.x * nrm);
    atomicAdd(dp + 1, m.y * nrm);
}

// ---------------------------------------------------------------- gather + self-loop + bias
__global__ void gce_gather(const long long* __restrict__ xidx, const float* __restrict__ acc,
                           const float* __restrict__ xw, const float* __restrict__ dinv,
                           const float* __restrict__ b, float* __restrict__ out) {
    int tid = blockIdx.x * blockDim.x + threadIdx.x;
    if (tid >= N_GATHER * OUT_DIM) return;
    const int i = tid >> 6;
    const int c = tid & 63;
    const int v = (int)xidx[i];
    const float di = dinv[v];
    out[tid] = acc[(size_t)v * OUT_DIM + c] + xw[(size_t)v * OUT_DIM + c] * di * di + b[c];
}

// ---------------------------------------------------------------- launch
extern "C" void kernel_launch(void* const* d_in, const int* in_sizes, int n_in,
                              void* d_out, int out_size, void* d_ws, size_t ws_size,
                              hipStream_t stream) {
    const float*     features = (const float*)d_in[0];          // [50000,128]
    const long long* edge     = (const long long*)d_in[1];      // [2,1600000]
    const long long* xidx     = (const long long*)d_in[2];      // [4096,2]
    const float*     W        = (const float*)d_in[3];          // [128,64]
    const float*     b        = (const float*)d_in[4];          // [64]
    float*           out      = (float*)d_out;                  // [4096,2,64]

    const long long* src = edge;
    const long long* dst = edge + N_EDGES;

    // workspace layout (all 16B-aligned): deg/dinv | xw | acc
    float* deg = (float*)d_ws;                         // 50000 floats
    float* xw  = deg + N_NODES;                        // 50000*64 floats
    float* acc = xw + (size_t)N_NODES * OUT_DIM;       // 50000*64 floats

    const int nAcc = N_NODES * OUT_DIM;                // 3,200,000

    gce_init  <<<(nAcc + 255) / 256, 256, 0, stream>>>(acc, deg, nAcc);
    gce_degree<<<(N_EDGES + 255) / 256, 256, 0, stream>>>(dst, deg);
    gce_dinv  <<<(N_NODES + 255) / 256, 256, 0, stream>>>(deg);
    gce_gemm  <<<(N_MTILES + 7) / 8, 256, 0, stream>>>(features, W, xw);
    {
        long long nThreads = (long long)N_EDGES * 32;
        gce_edges<<<(unsigned)((nThreads + 255) / 256), 256, 0, stream>>>(src, dst, xw, deg, acc);
    }
    gce_gather<<<(N_GATHER * OUT_DIM + 255) / 256, 256, 0, stream>>>(xidx, acc, xw, deg, b, out);
}